// MultiHeadAttention_88948772700366
// MI455X (gfx1250) — compile-verified
//
#include <hip/hip_runtime.h>

// MI455X / gfx1250 flash-attention, fp32 WMMA (V_WMMA_F32_16X16X4_F32) with
// Tensor-Data-Mover staging of the K/V tile into LDS (double-buffered).
// Shapes: T=S=1024, B=NHEAD=8, A=512, hd=64, N=B*NHEAD=64.
// Reference quirks: mask index is n % NHEAD (reshape aliasing), V == K.

typedef __attribute__((ext_vector_type(2))) float        v2f;
typedef __attribute__((ext_vector_type(8))) float        v8f;
typedef __attribute__((ext_vector_type(4))) unsigned int v4u;
typedef __attribute__((ext_vector_type(4))) int          v4i;
typedef __attribute__((ext_vector_type(8))) int          v8i;

#define WMMA_F32(a, b, c) \
  __builtin_amdgcn_wmma_f32_16x16x4_f32(false, (a), false, (b), (short)0, (c), false, false)

#define KPITCH 68   // 64 data DWORDs + 4 pad DWORDs per row (TDM pad_interval=5, pad_amount=3)

// Issue one TDM load of a 16-row x 64-float tile (row stride 4096 floats) into LDS,
// padded to a 68-float pitch.  D# layout per CDNA5 ISA ch.8.
__device__ __forceinline__ void tdm_load_tile(const float* gsrc, unsigned lds_off)
{
    unsigned long long ga = (unsigned long long)(uintptr_t)gsrc;
    v4u g0;
    g0[0] = 1u;                                            // count=1, user desc, no gather
    g0[1] = lds_off;                                       // lds_addr (bytes)
    g0[2] = (unsigned)ga;                                  // global_addr[31:0]
    g0[3] = ((unsigned)(ga >> 32) & 0x01FFFFFFu) | (2u << 30); // addr[56:32] | type=2
    v8i g1;
    g1[0] = (int)((2u << 16)      // data_size = 2 -> 4 bytes
                | (1u << 20)      // pad_enable
                | (5u << 22)      // pad_interval: 64 DWORDs
                | (3u << 25));    // pad_amount: 4 DWORDs
    g1[1] = (int)(64u << 16);     // tensor_dim0 = 64 elements
    g1[2] = (int)(16u << 16);     // tensor_dim1 = 16 rows
    g1[3] = (int)(64u << 16);     // tile_dim0 = 64
    g1[4] = 16;                   // tile_dim1 = 16, tile_dim2 = 0
    g1[5] = 4096;                 // tensor_dim0_stride = 4096 elements
    g1[6] = 0;
    g1[7] = 0;
    v4i z4 = {0, 0, 0, 0};
#if defined(__clang_major__) && __clang_major__ >= 23
    v8i z8 = {0, 0, 0, 0, 0, 0, 0, 0};
    __builtin_amdgcn_tensor_load_to_lds(g0, g1, z4, z4, z8, 0);
#else
    __builtin_amdgcn_tensor_load_to_lds(g0, g1, z4, z4, 0);
#endif
}

__global__ __launch_bounds__(256)
void MultiHeadAttention_88948772700366_kernel(const float* __restrict__ Q,
                                              const float* __restrict__ K,
                                              const unsigned char* __restrict__ Msk,
                                              float* __restrict__ Out)
{
    __shared__ float kstage[2][16 * KPITCH];    // double-buffered TDM K/V tile (2 x 4352 B)
    __shared__ float obuf[8 * 16 * 64];         // 32 KB: per-wave 16x64 O transpose buffer

    const int lane = threadIdx.x & 31;
    const int wave = threadIdx.x >> 5;
    const int lr   = lane & 15;                 // row/col within 16-group
    const int hlf  = lane >> 4;                 // which 16-lane half
    const int n    = blockIdx.x >> 3;           // head index 0..63
    const int tt   = (blockIdx.x & 7) * 128 + wave * 16;  // query-row tile base
    const int h    = n & 7;                     // mask slab index (reference quirk)

    // ---- Q tile (16t x 64d) in WMMA-B fragment layout -----------------------
    // B(4x16) chunk c: vgpr0 lane l = Q[t=tt+(l&15), d=4c+2*hlf]; vgpr1 = d+1.
    float qb[32];
    {
        const float* qrow = Q + (size_t)(tt + lr) * 4096 + n * 64 + 2 * hlf;
#pragma unroll
        for (int c = 0; c < 16; ++c) {
            v2f q2 = *(const v2f*)(qrow + 4 * c);
            qb[2 * c] = q2.x; qb[2 * c + 1] = q2.y;
        }
    }

    v8f o0 = {}, o1 = {}, o2 = {}, o3 = {};     // O^T tiles: d=0..15,16..31,32..47,48..63
    float mrun = -INFINITY, lrun = 0.0f;
    const float scale = 0.125f;                 // 1/sqrt(64)

    // 8 consecutive mask bytes per lane per s-tile (s0 + 8*hlf .. +7)
    const unsigned char* mbase =
        Msk + (size_t)h * 1024 * 1024 + (size_t)(tt + lr) * 1024 + 8 * hlf;

    const unsigned ldsK0 = (unsigned)(uintptr_t)(void*)&kstage[0][0];
    const unsigned ldsK1 = (unsigned)(uintptr_t)(void*)&kstage[1][0];

    // Prologue: stage tile 0.
    if (wave == 0) tdm_load_tile(K + (size_t)n * 64, ldsK0);

#pragma unroll 1
    for (int st = 0; st < 64; ++st) {
        // Issue next tile into the other buffer, then wait for the current one.
        if (wave == 0) {
            if (st < 63) {
                tdm_load_tile(K + (size_t)(st + 1) * 16 * 4096 + (size_t)n * 64,
                              ((st + 1) & 1) ? ldsK1 : ldsK0);
                __builtin_amdgcn_s_wait_tensorcnt((short)1);
            } else {
                __builtin_amdgcn_s_wait_tensorcnt((short)0);
            }
        }
        __syncthreads();                        // tile st visible to all waves
        const float* kb = kstage[st & 1];

        // ---- K tile fragments in WMMA-A layout (conflict-free b64 LDS reads)
        float ka[32];
        {
            const float* krow = kb + lr * KPITCH + 2 * hlf;
#pragma unroll
            for (int c = 0; c < 16; ++c) {
                v2f k2 = *(const v2f*)(krow + 4 * c);
                ka[2 * c] = k2.x; ka[2 * c + 1] = k2.y;
            }
        }
        // ---- V^T fragments for PV: A(16d x 4s) chunks. vv[c4][kk][dt]
        float vv[4][2][4];
#pragma unroll
        for (int c4 = 0; c4 < 4; ++c4)
#pragma unroll
            for (int kk = 0; kk < 2; ++kk) {
                const float* vrow = kb + (4 * c4 + 2 * hlf + kk) * KPITCH + lr;
#pragma unroll
                for (int dt = 0; dt < 4; ++dt) vv[c4][kk][dt] = vrow[16 * dt];
            }

        // ---- S^T(16s x 16t) = K_tile(16x64) x Q^T(64x16): 16 fp32 WMMAs ----
        v8f acc = {};
#pragma unroll
        for (int c = 0; c < 16; ++c) {
            v2f a = { ka[2 * c], ka[2 * c + 1] };
            v2f b = { qb[2 * c], qb[2 * c + 1] };
            acc = WMMA_F32(a, b, acc);
        }

        // ---- scale + mask (D vgpr r <-> s = s0 + r + 8*hlf; lane <-> t) -----
        unsigned long long m8 = *(const unsigned long long*)(mbase + st * 16);
        float sc[8];
#pragma unroll
        for (int r = 0; r < 8; ++r) {
            bool keep = ((m8 >> (8 * r)) & 0xffull) != 0;
            sc[r] = keep ? acc[r] * scale : -1e30f;
        }

        // ---- online softmax over s (per t-column = per lane) ----------------
        float mt = sc[0];
#pragma unroll
        for (int r = 1; r < 8; ++r) mt = fmaxf(mt, sc[r]);
        mt = fmaxf(mt, __shfl_xor(mt, 16, 32));
        const float mnew = fmaxf(mrun, mt);

        float p[8], ps = 0.0f;
#pragma unroll
        for (int r = 0; r < 8; ++r) { p[r] = __expf(sc[r] - mnew); ps += p[r]; }
        ps += __shfl_xor(ps, 16, 32);

        const float alpha = __expf(mrun - mnew);
        lrun = lrun * alpha + ps;
        mrun = mnew;
#pragma unroll
        for (int r = 0; r < 8; ++r) {
            o0[r] *= alpha; o1[r] *= alpha; o2[r] *= alpha; o3[r] *= alpha;
        }

        // ---- cross-half copies of P^T rows (xor-16 swap) --------------------
        float px[8];
#pragma unroll
        for (int r = 0; r < 8; ++r) px[r] = __shfl_xor(p[r], 16, 32);

        // ---- O^T += V^T x P^T : 16 fp32 WMMAs -------------------------------
#pragma unroll
        for (int c4 = 0; c4 < 4; ++c4) {
            float b0, b1;   // B(4x16): vgpr0 lane l = P^T[s=4*c4+2*hlf, t=l&15], vgpr1 = s+1
            if (c4 == 0)      { b0 = hlf ? px[2] : p[0];  b1 = hlf ? px[3] : p[1]; }
            else if (c4 == 1) { b0 = hlf ? px[6] : p[4];  b1 = hlf ? px[7] : p[5]; }
            else if (c4 == 2) { b0 = hlf ? p[2] : px[0];  b1 = hlf ? p[3] : px[1]; }
            else              { b0 = hlf ? p[6] : px[4];  b1 = hlf ? p[7] : px[5]; }
            v2f bv = { b0, b1 };
            v2f a0 = { vv[c4][0][0], vv[c4][1][0] };  o0 = WMMA_F32(a0, bv, o0);
            v2f a1 = { vv[c4][0][1], vv[c4][1][1] };  o1 = WMMA_F32(a1, bv, o1);
            v2f a2 = { vv[c4][0][2], vv[c4][1][2] };  o2 = WMMA_F32(a2, bv, o2);
            v2f a3 = { vv[c4][0][3], vv[c4][1][3] };  o3 = WMMA_F32(a3, bv, o3);
        }
        __syncthreads();                        // all reads of buf[st&1] done before reuse
    }

    // ---- normalize, transpose through LDS, coalesced b128 stores ------------
    const float inv = 1.0f / lrun;              // per lane: t = lr (both halves equal)
    float* myLds = obuf + wave * 1024;
#pragma unroll
    for (int r = 0; r < 8; ++r) {
        myLds[lr * 64 +  0 + r + 8 * hlf] = o0[r] * inv;
        myLds[lr * 64 + 16 + r + 8 * hlf] = o1[r] * inv;
        myLds[lr * 64 + 32 + r + 8 * hlf] = o2[r] * inv;
        myLds[lr * 64 + 48 + r + 8 * hlf] = o3[r] * inv;
    }
    __syncthreads();

    const int t2   = lane >> 1;                 // 2 lanes per output row
    const int colb = (lane & 1) * 32;
    const float4* src = (const float4*)(myLds + t2 * 64 + colb);
    float4* dst = (float4*)(Out + (size_t)(tt + t2) * 4096 + n * 64 + colb);
#pragma unroll
    for (int j = 0; j < 8; ++j) dst[j] = src[j];
}

extern "C" void kernel_launch(void* const* d_in, const int* in_sizes, int n_in,
                              void* d_out, int out_size, void* d_ws, size_t ws_size,
                              hipStream_t stream) {
    const float* Q = (const float*)d_in[0];          // [1024, 8, 512] fp32
    const float* K = (const float*)d_in[1];          // [1024, 8, 512] fp32
    const unsigned char* M = (const unsigned char*)d_in[2];  // [8,1024,1024] bool bytes
    float* Out = (float*)d_out;                      // [1024, 8, 512] fp32

    dim3 grid(64 * 8);     // 64 heads x 8 query-row tiles of 128
    dim3 block(256);       // 8 waves; each wave owns a 16-row flash-attention tile
    hipLaunchKernelGGL(MultiHeadAttention_88948772700366_kernel,
                       grid, block, 0, stream, Q, K, M, Out);
}